// BiMambaBlock_85109071937986
// MI455X (gfx1250) — compile-verified
//
#include <hip/hip_runtime.h>
#include <hip/hip_bf16.h>
#include <math.h>

// ---------------- problem constants ----------------
#define D_MODEL 256
#define D_STATE 16
#define BATCH   4
#define SEQ     16384
#define MROWS   (BATCH * SEQ)      // 65536 rows
#define CLEN    128                // scan chunk length
#define CHUNKS  (SEQ / CLEN)       // 128 chunks
#define LN_EPS  1e-5f

typedef float v2f __attribute__((ext_vector_type(2)));
typedef float v8f __attribute__((ext_vector_type(8)));
typedef unsigned int u32x4 __attribute__((ext_vector_type(4)));
typedef int i32x4 __attribute__((ext_vector_type(4)));
typedef int i32x8 __attribute__((ext_vector_type(8)));

// CDNA5 fp32 WMMA: D(16x16,f32) = A(16x4,f32) * B(4x16,f32) + C
// A frag: lane l -> row M=l%16, holds K = 2*(l/16)+{0,1}
// B frag: lane l -> col N=l%16, holds K = 2*(l/16)+{0,1}
// C/D:    lane l, vgpr r -> M = r + 8*(l/16), N = l%16
__device__ __forceinline__ v8f wmma_f32(v2f a, v2f b, v8f c) {
  return __builtin_amdgcn_wmma_f32_16x16x4_f32(false, a, false, b, (short)0, c,
                                               false, false);
}

// ---------------------------------------------------------------------
// Tensor Data Mover: DMA a 2D fp32 tile (tile_w x tile_h, row stride
// row_stride elems) from global memory into LDS at lds_byte_addr, with
// hardware LDS padding (pad_amount DWORDs inserted after every
// 2^(pad_interval_code+1) DWORDs stored).  D# layout per CDNA5 ISA §8.
// Toolchain here exposes the 6-arg builtin:
//   (u32x4 g0, i32x8 g1, i32x4 g2, i32x4 g3, i32x8 extra, i32 cpol)
// ---------------------------------------------------------------------
__device__ __forceinline__ void tdm_load_tile_2d(
    unsigned lds_byte_addr, const float* gsrc, unsigned tile_w,
    unsigned tile_h, unsigned row_stride, unsigned pad_interval_code,
    unsigned pad_amount_code) {
  unsigned long long ga = (unsigned long long)(const void*)gsrc;
  u32x4 g0;
  g0.x = 1u;                                  // count=1, user descriptor
  g0.y = lds_byte_addr;                       // lds_addr  [63:32]
  g0.z = (unsigned)ga;                        // global_addr[95:64]
  g0.w = (unsigned)((ga >> 32) & 0x01FFFFFFu) // global_addr[120:96]
         | (2u << 30);                        // type=2 ("image") [127:126]

  i32x8 g1;
  g1[0] = (int)((2u << 16)                    // data_size = 4 bytes
                | (1u << 20)                  // pad_enable
                | (pad_interval_code << 22)   // pad interval
                | (pad_amount_code << 25));   // pad amount (DWORDs-1)
  g1[1] = (int)((tile_w & 0xFFFFu) << 16);    // tensor_dim0[15:0] @ [63:48]
  g1[2] = (int)(((tile_w >> 16) & 0xFFFFu)    // tensor_dim0[31:16]
                | ((tile_h & 0xFFFFu) << 16));// tensor_dim1[15:0]
  g1[3] = (int)(((tile_h >> 16) & 0xFFFFu)    // tensor_dim1[31:16]
                | ((tile_w & 0xFFFFu) << 16));// tile_dim0 @ [127:112]
  g1[4] = (int)(tile_h & 0xFFFFu);            // tile_dim1 ; tile_dim2 = 0
  g1[5] = (int)row_stride;                    // tensor_dim0_stride[31:0]
  g1[6] = 0;                                  // stride hi / dim1_stride (2D)
  g1[7] = 0;
  i32x4 z4 = {0, 0, 0, 0};
  i32x8 z8 = {0, 0, 0, 0, 0, 0, 0, 0};
  __builtin_amdgcn_tensor_load_to_lds(g0, g1, z4, z4, z8, 0);
}

// =====================================================================
// Kernel 1: value = X @ W_proj[:,256:512] + b_proj[256:512]
//   X: (M,256) row-major, W_proj: (256,512) row-major, Y: (M,256)
//   grid = M/16 row panels, 256 threads = 8 waves; wave w does N cols
//   [32w,32w+32).  A panel DMA'd by the TDM into LDS with +4 DWORD row
//   padding (stride 260 => 260%64==4 -> conflict-free 8B frag loads).
// =====================================================================
#define LDA 260
__global__ __launch_bounds__(256) void gemm_value_kernel(
    const float* __restrict__ X, const float* __restrict__ Wp,
    const float* __restrict__ bp, float* __restrict__ Y) {
  __shared__ float As[16 * LDA];
  const int panel = blockIdx.x;
  const int tid = threadIdx.x;
  const int lane = tid & 31;
  const int wave = tid >> 5;

  if (tid < 32) { // one TDM descriptor stages the whole 16x256 panel
    tdm_load_tile_2d((unsigned)(size_t)As,
                     X + (size_t)panel * 16 * D_MODEL,
                     /*tile_w=*/256, /*tile_h=*/16, /*row_stride=*/256,
                     /*pad_interval: 2^(7+1)=256 DW*/ 7u,
                     /*pad_amount: 4 DW*/ 3u);
    __builtin_amdgcn_s_wait_tensorcnt(0);
  }
  __syncthreads();

  const int nlane = lane & 15;
  const int m = lane & 15;
  const int kk = (lane >> 4) * 2;
  const int n0 = wave * 32;

  v8f acc0 = {};
  v8f acc1 = {};
  for (int k0 = 0; k0 < 256; k0 += 4) {
    if ((k0 & 63) == 0 && k0 + 64 < 256)
      __builtin_prefetch(Wp + (size_t)(k0 + 64 + kk) * 512 + 256 + n0 + nlane, 0, 1);
    v2f a;
    a.x = As[m * LDA + k0 + kk];
    a.y = As[m * LDA + k0 + kk + 1];
    const float* w0 = Wp + (size_t)(k0 + kk) * 512 + 256;
    v2f b0, b1;
    b0.x = w0[n0 + nlane];
    b0.y = w0[512 + n0 + nlane];
    b1.x = w0[16 + n0 + nlane];
    b1.y = w0[512 + 16 + n0 + nlane];
    acc0 = wmma_f32(a, b0, acc0);
    acc1 = wmma_f32(a, b1, acc1);
  }

  const int mbase = panel * 16 + (lane >> 4) * 8;
#pragma unroll
  for (int r = 0; r < 8; ++r) {
    Y[(size_t)(mbase + r) * D_MODEL + n0 + nlane] =
        acc0[r] + bp[256 + n0 + nlane];
    Y[(size_t)(mbase + r) * D_MODEL + n0 + 16 + nlane] =
        acc1[r] + bp[256 + 16 + nlane + n0];
  }
}

// =====================================================================
// Kernel 2: G = sigmoid(V @ W_gate + b_gate)   V:(M,256) Wg:(256,16)
//   grid = M/128, 8 waves each own a 16-row x 16-col tile.
// =====================================================================
__global__ __launch_bounds__(256) void gemm_gate_kernel(
    const float* __restrict__ V, const float* __restrict__ Wg,
    const float* __restrict__ bg, float* __restrict__ G) {
  const int tid = threadIdx.x;
  const int lane = tid & 31;
  const int wave = tid >> 5;
  const int rowbase = blockIdx.x * 128 + wave * 16;
  const int m = rowbase + (lane & 15);
  const int n = lane & 15;
  const int kk = (lane >> 4) * 2;

  v8f acc = {};
  for (int k0 = 0; k0 < 256; k0 += 4) {
    v2f a, b;
    const float* vp = V + (size_t)m * D_MODEL + k0 + kk;
    a.x = vp[0];
    a.y = vp[1];
    b.x = Wg[(k0 + kk) * D_STATE + n];
    b.y = Wg[(k0 + kk + 1) * D_STATE + n];
    acc = wmma_f32(a, b, acc);
  }
  const int mb = rowbase + (lane >> 4) * 8;
#pragma unroll
  for (int r = 0; r < 8; ++r) {
    float x = acc[r] + bg[n];
    G[(size_t)(mb + r) * D_STATE + n] = 1.0f / (1.0f + __expf(-x));
  }
}

// =====================================================================
// Scan phase A: per (b,chunk): zero-init partial state + chunk decay prod
// =====================================================================
__global__ __launch_bounds__(256) void scan_phaseA_kernel(
    const float* __restrict__ G, const float* __restrict__ V,
    float* __restrict__ cfinal, float* __restrict__ Pout, int reverse) {
  const int blk = blockIdx.x;
  const int b = blk / CHUNKS;
  const int c = blk % CHUNKS;
  const int d = threadIdx.x;

  float cs[16], P[16];
#pragma unroll
  for (int s = 0; s < 16; ++s) { cs[s] = 0.0f; P[s] = 1.0f; }

  for (int i = 0; i < CLEN; ++i) {
    const int step = c * CLEN + i;
    const int tau = reverse ? (SEQ - 1 - step) : step;
    const size_t row = (size_t)b * SEQ + tau;
    const float4* gp = (const float4*)(G + row * D_STATE);
    float4 q0 = gp[0], q1 = gp[1], q2 = gp[2], q3 = gp[3];
    float gv[16] = {q0.x, q0.y, q0.z, q0.w, q1.x, q1.y, q1.z, q1.w,
                    q2.x, q2.y, q2.z, q2.w, q3.x, q3.y, q3.z, q3.w};
    const float vv = V[row * D_MODEL + d];
#pragma unroll
    for (int s = 0; s < 16; ++s) {
      cs[s] = gv[s] * cs[s] + (1.0f - gv[s]) * vv;
      P[s] *= gv[s];
    }
  }
  const size_t base = (size_t)blk * D_STATE * D_MODEL;
#pragma unroll
  for (int s = 0; s < 16; ++s) cfinal[base + s * D_MODEL + d] = cs[s];
  if (d < 16) Pout[(size_t)blk * D_STATE + d] = P[d];
}

// =====================================================================
// Scan phase B: sequential chunk combine (128 steps). grid = BATCH.
// =====================================================================
__global__ __launch_bounds__(256) void scan_phaseB_kernel(
    const float* __restrict__ cfinal, const float* __restrict__ P,
    float* __restrict__ cinit) {
  const int b = blockIdx.x;
  const int d = threadIdx.x;
  float cs[16];
#pragma unroll
  for (int s = 0; s < 16; ++s) cs[s] = 0.0f;
  for (int c = 0; c < CHUNKS; ++c) {
    const size_t base = (size_t)(b * CHUNKS + c) * D_STATE * D_MODEL;
    const float* Pp = P + (size_t)(b * CHUNKS + c) * D_STATE;
#pragma unroll
    for (int s = 0; s < 16; ++s) {
      cinit[base + s * D_MODEL + d] = cs[s];
      cs[s] = Pp[s] * cs[s] + cfinal[base + s * D_MODEL + d];
    }
  }
}

// =====================================================================
// Scan phase C: replay chunk from true init, emit outputs into the
//   concatenated (M,512) combined buffer at column offset outOff.
// =====================================================================
__global__ __launch_bounds__(256) void scan_phaseC_kernel(
    const float* __restrict__ G, const float* __restrict__ V,
    const float* __restrict__ cinit, const float* __restrict__ Amat,
    float* __restrict__ out, int reverse, int outOff) {
  const int blk = blockIdx.x;
  const int b = blk / CHUNKS;
  const int c = blk % CHUNKS;
  const int d = threadIdx.x;

  float cs[16], av[16];
  const size_t base = (size_t)blk * D_STATE * D_MODEL;
#pragma unroll
  for (int s = 0; s < 16; ++s) {
    cs[s] = cinit[base + s * D_MODEL + d];
    av[s] = Amat[s * D_MODEL + d];
  }

  for (int i = 0; i < CLEN; ++i) {
    const int step = c * CLEN + i;
    const int tau = reverse ? (SEQ - 1 - step) : step;
    const size_t row = (size_t)b * SEQ + tau;
    const float4* gp = (const float4*)(G + row * D_STATE);
    float4 q0 = gp[0], q1 = gp[1], q2 = gp[2], q3 = gp[3];
    float gv[16] = {q0.x, q0.y, q0.z, q0.w, q1.x, q1.y, q1.z, q1.w,
                    q2.x, q2.y, q2.z, q2.w, q3.x, q3.y, q3.z, q3.w};
    const float vv = V[row * D_MODEL + d];
    float o = 0.0f;
#pragma unroll
    for (int s = 0; s < 16; ++s) {
      cs[s] = gv[s] * cs[s] + (1.0f - gv[s]) * vv;
      o += av[s] * gv[s] * cs[s];
    }
    out[row * 512 + outOff + d] = o;
  }
}

// =====================================================================
// Kernel 6: out = LN(combined @ W_out + b_out)*ln_g + ln_b
//   combined:(M,512), W_out:(512,256). grid = M/16 row panels.
//   A panel DMA'd by TDM: pad 2 DWORDs after every 256 stored ->
//   LDS row stride 516 (516%64==4, conflict-free) with a 2-DWORD gap
//   at mid-row; fragment loads add +2 for k >= 256.
// =====================================================================
#define LDK 516
__global__ __launch_bounds__(256) void gemm_out_ln_kernel(
    const float* __restrict__ Cm, const float* __restrict__ Wo,
    const float* __restrict__ bo, const float* __restrict__ lng,
    const float* __restrict__ lnb, float* __restrict__ out) {
  __shared__ float As[16 * LDK];
  __shared__ float Cs[16 * 256];
  __shared__ float red[2 * 256 + 32];
  const int panel = blockIdx.x;
  const int tid = threadIdx.x;
  const int lane = tid & 31;
  const int wave = tid >> 5;

  if (tid < 32) { // TDM stage of the 16x512 panel
    tdm_load_tile_2d((unsigned)(size_t)As,
                     Cm + (size_t)panel * 16 * 512,
                     /*tile_w=*/512, /*tile_h=*/16, /*row_stride=*/512,
                     /*pad_interval: 256 DW*/ 7u,
                     /*pad_amount: 2 DW*/ 1u);
    __builtin_amdgcn_s_wait_tensorcnt(0);
  }
  __syncthreads();

  const int nlane = lane & 15;
  const int m = lane & 15;
  const int kk = (lane >> 4) * 2;
  const int n0 = wave * 32;

  v8f acc0 = {};
  v8f acc1 = {};
  for (int k0 = 0; k0 < 512; k0 += 4) {
    if ((k0 & 63) == 0 && k0 + 64 < 512)
      __builtin_prefetch(Wo + (size_t)(k0 + 64 + kk) * 256 + n0 + nlane, 0, 1);
    const int koff = (k0 & 256) ? 2 : 0;  // skip mid-row pad gap
    v2f a;
    a.x = As[m * LDK + k0 + kk + koff];
    a.y = As[m * LDK + k0 + kk + koff + 1];
    const float* w0 = Wo + (size_t)(k0 + kk) * 256;
    v2f b0, b1;
    b0.x = w0[n0 + nlane];
    b0.y = w0[256 + n0 + nlane];
    b1.x = w0[16 + n0 + nlane];
    b1.y = w0[256 + 16 + n0 + nlane];
    acc0 = wmma_f32(a, b0, acc0);
    acc1 = wmma_f32(a, b1, acc1);
  }

  { // deposit C tiles (+bias) into LDS for row-wise LayerNorm
    const int mh = (lane >> 4) * 8;
#pragma unroll
    for (int r = 0; r < 8; ++r) {
      Cs[(mh + r) * 256 + n0 + nlane] = acc0[r] + bo[n0 + nlane];
      Cs[(mh + r) * 256 + n0 + 16 + nlane] = acc1[r] + bo[n0 + 16 + nlane];
    }
  }
  __syncthreads();

  { // partial sums: thread covers (row = tid/16, 16-col segment tid%16)
    const int row = tid >> 4;
    const int seg = tid & 15;
    float s = 0.0f, q = 0.0f;
#pragma unroll
    for (int j = 0; j < 16; ++j) {
      float x = Cs[row * 256 + seg * 16 + j];
      s += x;
      q += x * x;
    }
    red[row * 16 + seg] = s;
    red[256 + row * 16 + seg] = q;
  }
  __syncthreads();
  if (tid < 16) {
    float ss = 0.0f, qq = 0.0f;
#pragma unroll
    for (int j = 0; j < 16; ++j) {
      ss += red[tid * 16 + j];
      qq += red[256 + tid * 16 + j];
    }
    const float mu = ss * (1.0f / 256.0f);
    const float var = qq * (1.0f / 256.0f) - mu * mu;
    red[512 + tid] = mu;
    red[528 + tid] = rsqrtf(var + LN_EPS);
  }
  __syncthreads();
  {
    const int col = tid;
    const float gsc = lng[col];
    const float bsc = lnb[col];
#pragma unroll
    for (int r = 0; r < 16; ++r) {
      out[((size_t)panel * 16 + r) * 256 + col] =
          (Cs[r * 256 + col] - red[512 + r]) * red[528 + r] * gsc + bsc;
    }
  }
}

// =====================================================================
// Host-side orchestration
// =====================================================================
extern "C" void kernel_launch(void* const* d_in, const int* in_sizes, int n_in,
                              void* d_out, int out_size, void* d_ws,
                              size_t ws_size, hipStream_t stream) {
  (void)in_sizes; (void)n_in; (void)out_size; (void)ws_size;
  const float* x       = (const float*)d_in[0];
  const float* W_fproj = (const float*)d_in[1];
  const float* b_fproj = (const float*)d_in[2];
  const float* A_f     = (const float*)d_in[3];
  const float* W_fgate = (const float*)d_in[4];
  const float* b_fgate = (const float*)d_in[5];
  const float* W_bproj = (const float*)d_in[6];
  const float* b_bproj = (const float*)d_in[7];
  const float* A_b     = (const float*)d_in[8];
  const float* W_bgate = (const float*)d_in[9];
  const float* b_bgate = (const float*)d_in[10];
  const float* W_out   = (const float*)d_in[11];
  const float* b_out   = (const float*)d_in[12];
  const float* ln_g    = (const float*)d_in[13];
  const float* ln_b    = (const float*)d_in[14];
  float* out = (float*)d_out;

  // workspace carve-up
  float* w = (float*)d_ws;
  size_t off = 0;
  float* value_f = w + off;  off += (size_t)MROWS * D_MODEL;
  float* value_b = w + off;  off += (size_t)MROWS * D_MODEL;
  float* g_f     = w + off;  off += (size_t)MROWS * D_STATE;
  float* g_b     = w + off;  off += (size_t)MROWS * D_STATE;
  float* combined= w + off;  off += (size_t)MROWS * 512;
  float* cfin_f  = w + off;  off += (size_t)BATCH * CHUNKS * D_STATE * D_MODEL;
  float* cfin_b  = w + off;  off += (size_t)BATCH * CHUNKS * D_STATE * D_MODEL;
  float* cini_f  = w + off;  off += (size_t)BATCH * CHUNKS * D_STATE * D_MODEL;
  float* cini_b  = w + off;  off += (size_t)BATCH * CHUNKS * D_STATE * D_MODEL;
  float* P_f     = w + off;  off += (size_t)BATCH * CHUNKS * D_STATE;
  float* P_b     = w + off;  off += (size_t)BATCH * CHUNKS * D_STATE;

  const dim3 blk(256);

  // 1) value projections (only second half of W_proj is live)
  gemm_value_kernel<<<MROWS / 16, blk, 0, stream>>>(x, W_fproj, b_fproj, value_f);
  gemm_value_kernel<<<MROWS / 16, blk, 0, stream>>>(x, W_bproj, b_bproj, value_b);

  // 2) gates
  gemm_gate_kernel<<<MROWS / 128, blk, 0, stream>>>(value_f, W_fgate, b_fgate, g_f);
  gemm_gate_kernel<<<MROWS / 128, blk, 0, stream>>>(value_b, W_bgate, b_bgate, g_b);

  // 3) chunked scans (forward and reversed)
  scan_phaseA_kernel<<<BATCH * CHUNKS, blk, 0, stream>>>(g_f, value_f, cfin_f, P_f, 0);
  scan_phaseA_kernel<<<BATCH * CHUNKS, blk, 0, stream>>>(g_b, value_b, cfin_b, P_b, 1);
  scan_phaseB_kernel<<<BATCH, blk, 0, stream>>>(cfin_f, P_f, cini_f);
  scan_phaseB_kernel<<<BATCH, blk, 0, stream>>>(cfin_b, P_b, cini_b);
  scan_phaseC_kernel<<<BATCH * CHUNKS, blk, 0, stream>>>(g_f, value_f, cini_f, A_f,
                                                         combined, 0, 0);
  scan_phaseC_kernel<<<BATCH * CHUNKS, blk, 0, stream>>>(g_b, value_b, cini_b, A_b,
                                                         combined, 1, 256);

  // 4) output projection + LayerNorm
  gemm_out_ln_kernel<<<MROWS / 16, blk, 0, stream>>>(combined, W_out, b_out,
                                                     ln_g, ln_b, out);
}